// PointNetSetAbstraction_33217277067288
// MI455X (gfx1250) — compile-verified
//
#include <hip/hip_runtime.h>
#include <hip/hip_bf16.h>

// ---------------- problem constants ----------------
constexpr int BB      = 16;
constexpr int NN      = 4096;
constexpr int CC      = 64;
constexpr int NPOINT  = 1024;
constexpr int NSAMPLE = 32;
constexpr float R2    = 0.2f * 0.2f;

typedef __attribute__((ext_vector_type(2))) float v2f;
typedef __attribute__((ext_vector_type(8))) float v8f;

// =====================================================================
// Kernel 1: farthest point sampling. One block per batch, 256 threads.
// Point coords + running min-dist live in registers (16 pts/thread);
// single __syncthreads per iteration via parity-double-buffered partials.
// =====================================================================
__global__ __launch_bounds__(256) void fps_kernel(const float* __restrict__ xyz,
                                                  int* __restrict__ fps_idx,
                                                  float* __restrict__ new_xyz) {
  __shared__ float sxyz[NN * 3];          // 48 KB (centroid lookup only)
  __shared__ float swv[2][8];             // double-buffered wave partials
  __shared__ int   swi[2][8];

  const int b   = blockIdx.x;
  const int tid = threadIdx.x;
  const float* xb = xyz + (size_t)b * NN * 3;

  for (int i = tid; i < NN * 3; i += 256) sxyz[i] = xb[i];
  __syncthreads();

  float px[16], py[16], pz[16], dist[16];
#pragma unroll
  for (int k = 0; k < 16; ++k) {
    const int p = tid + k * 256;
    px[k] = sxyz[p * 3 + 0];
    py[k] = sxyz[p * 3 + 1];
    pz[k] = sxyz[p * 3 + 2];
    dist[k] = 1e10f;
  }

  const int lane = tid & 31;
  const int wid  = tid >> 5;
  int far = 0;

  for (int s = 0; s < NPOINT; ++s) {
    if (tid == 0) {
      fps_idx[b * NPOINT + s] = far;
      new_xyz[(size_t)(b * NPOINT + s) * 3 + 0] = sxyz[far * 3 + 0];
      new_xyz[(size_t)(b * NPOINT + s) * 3 + 1] = sxyz[far * 3 + 1];
      new_xyz[(size_t)(b * NPOINT + s) * 3 + 2] = sxyz[far * 3 + 2];
    }
    const float cx = sxyz[far * 3 + 0];   // broadcast LDS reads
    const float cy = sxyz[far * 3 + 1];
    const float cz = sxyz[far * 3 + 2];

    float bestv = -1.0f;
    int   besti = 0;
#pragma unroll
    for (int k = 0; k < 16; ++k) {
      const float dx = px[k] - cx;
      const float dy = py[k] - cy;
      const float dz = pz[k] - cz;
      const float d  = dx * dx + dy * dy + dz * dz;
      dist[k] = fminf(dist[k], d);
      if (dist[k] > bestv) { bestv = dist[k]; besti = tid + k * 256; }
    }
    // wave32 argmax reduction, first-index tie-break (matches jnp.argmax)
#pragma unroll
    for (int off = 16; off > 0; off >>= 1) {
      const float ov = __shfl_xor(bestv, off, 32);
      const int   oi = __shfl_xor(besti, off, 32);
      if (ov > bestv || (ov == bestv && oi < besti)) { bestv = ov; besti = oi; }
    }
    const int buf = s & 1;
    if (lane == 0) { swv[buf][wid] = bestv; swi[buf][wid] = besti; }
    __syncthreads();
    // every thread combines the 8 partials (uniform result, no broadcast stage)
    float bv = swv[buf][0]; int bi = swi[buf][0];
#pragma unroll
    for (int w = 1; w < 8; ++w) {
      const float wv = swv[buf][w]; const int wi = swi[buf][w];
      if (wv > bv || (wv == bv && wi < bi)) { bv = wv; bi = wi; }
    }
    far = bi;
  }
}

// =====================================================================
// Kernel 2: ball query. One wave32 per centroid; ballot + prefix popcount
// collects the first NSAMPLE in-range indices in ascending order, padded
// with the first hit (exactly the reference's sort/take/where semantics).
// =====================================================================
__global__ __launch_bounds__(256) void ball_kernel(const float* __restrict__ xyz,
                                                   const float* __restrict__ new_xyz,
                                                   const int* __restrict__ fps_idx,
                                                   int* __restrict__ idx_ws,
                                                   float* __restrict__ idx_out) {
  __shared__ int sidx[8][NSAMPLE];
  const int lane = threadIdx.x & 31;
  const int wloc = threadIdx.x >> 5;
  const int gw   = blockIdx.x * 8 + wloc;          // centroid id, 0..16383
  if (gw >= BB * NPOINT) return;

  const int b = gw / NPOINT;
  const float cx = new_xyz[gw * 3 + 0];
  const float cy = new_xyz[gw * 3 + 1];
  const float cz = new_xyz[gw * 3 + 2];
  const float* xb = xyz + (size_t)b * NN * 3;

  int* row = sidx[wloc];
  int found = 0;
  for (int base = 0; base < NN && found < NSAMPLE; base += 32) {
    const int p = base + lane;
    const float dx = xb[p * 3 + 0] - cx;
    const float dy = xb[p * 3 + 1] - cy;
    const float dz = xb[p * 3 + 2] - cz;
    const bool in = (dx * dx + dy * dy + dz * dz) <= R2;
    const unsigned m32 = (unsigned)__ballot(in);
    const int pre  = __popc(m32 & ((1u << lane) - 1u));
    const int slot = found + pre;
    if (in && slot < NSAMPLE) row[slot] = p;
    found += __popc(m32);
  }
  if (found == 0) { row[0] = fps_idx[gw]; found = 1; }   // unreachable in practice
  if (found > NSAMPLE) found = NSAMPLE;

  const int first = row[0];
  const int v = (lane < found) ? row[lane] : first;
  idx_ws[gw * NSAMPLE + lane]  = v;
  idx_out[gw * NSAMPLE + lane] = (float)v;
}

// =====================================================================
// Kernel 3: fused gather + 3-layer MLP (FP32 WMMA 16x16x4) + max-pool.
// 256 threads = 8 waves; 4 centroids (128 rows) per block; each wave
// owns one 16-row M-tile for the whole pipeline (no cross-wave barriers
// between layers — activation rows are wave-private in LDS).
// =====================================================================
constexpr int LDK = 68;   // padded K for layer 1 (3 + 64 + 1 zero pad)

__global__ __launch_bounds__(256) void mlp_kernel(const float* __restrict__ xyz,
                                                  const float* __restrict__ pts,
                                                  const int* __restrict__ idx_ws,
                                                  const float* __restrict__ new_xyz,
                                                  const float* __restrict__ W1,
                                                  const float* __restrict__ b1,
                                                  const float* __restrict__ W2,
                                                  const float* __restrict__ b2,
                                                  const float* __restrict__ W3,
                                                  const float* __restrict__ b3,
                                                  float* __restrict__ out_pts) {
  __shared__ float X[128 * LDK];    // 34,816 B : input tile, then reused for H1/H2
  __shared__ float pmax[8 * 128];   // 4 KB    : per-wave column maxima

  const int tid = threadIdx.x;
  const int s0  = blockIdx.x * 4;   // first of 4 centroids

  // ---- gather: rows = [dx,dy,dz, f0..f63, 0] -------------------------
  for (int e = tid; e < 128 * LDK; e += 256) {
    const int r = e / LDK, c = e - r * LDK;
    const int g = r >> 5, t = r & 31;
    const int sg = s0 + g;
    const int b  = sg >> 10;                   // / NPOINT
    const int j  = idx_ws[sg * NSAMPLE + t];
    float v;
    if (c < 3)       v = xyz[((size_t)b * NN + j) * 3 + c] - new_xyz[sg * 3 + c];
    else if (c < 67) v = pts[((size_t)b * NN + j) * CC + (c - 3)];
    else             v = 0.0f;
    X[e] = v;
  }
  __syncthreads();

  const int wave = tid >> 5, lane = tid & 31;
  const int m0 = wave * 16;
  const int am = lane & 15;                 // A row / B col / C col
  const int ak = (lane >> 4) * 2;           // K sub-offset for A and B frags
  const int chi = (lane >> 4) * 8;          // C/D row offset

  auto Afrag = [&](int k0) -> v2f {
    const int base = (m0 + am) * LDK + k0 + ak;
    v2f a; a.x = X[base]; a.y = X[base + 1]; return a;
  };
  auto Bfrag = [&](const float* W, int ldn, int k0, int n0, int kmax) -> v2f {
    const int kk = k0 + ak;
    const int r0 = (kk     < kmax) ? kk     : kmax;
    const int r1 = (kk + 1 < kmax) ? kk + 1 : kmax;
    v2f bf; bf.x = W[r0 * ldn + n0 + am]; bf.y = W[r1 * ldn + n0 + am]; return bf;
  };

  // ---- layer 1: (128x68) x (68x64) ----------------------------------
  {
    v8f acc[4] = {};
    for (int k0 = 0; k0 < 68; k0 += 4) {
      const v2f a = Afrag(k0);
#pragma unroll
      for (int nt = 0; nt < 4; ++nt) {
        const v2f bf = Bfrag(W1, 64, k0, nt * 16, 66);  // clamp: X col 67 is 0
        acc[nt] = __builtin_amdgcn_wmma_f32_16x16x4_f32(false, a, false, bf,
                                                        (short)0, acc[nt], false, false);
      }
    }
#pragma unroll
    for (int nt = 0; nt < 4; ++nt) {
      const float bb = b1[nt * 16 + am];
#pragma unroll
      for (int r = 0; r < 8; ++r) {
        X[(m0 + r + chi) * LDK + nt * 16 + am] = fmaxf(acc[nt][r] + bb, 0.0f);
      }
    }
  }

  // ---- layer 2: (128x64) x (64x64) ----------------------------------
  {
    v8f acc[4] = {};
    for (int k0 = 0; k0 < 64; k0 += 4) {
      const v2f a = Afrag(k0);
#pragma unroll
      for (int nt = 0; nt < 4; ++nt) {
        const v2f bf = Bfrag(W2, 64, k0, nt * 16, 63);
        acc[nt] = __builtin_amdgcn_wmma_f32_16x16x4_f32(false, a, false, bf,
                                                        (short)0, acc[nt], false, false);
      }
    }
#pragma unroll
    for (int nt = 0; nt < 4; ++nt) {
      const float bb = b2[nt * 16 + am];
#pragma unroll
      for (int r = 0; r < 8; ++r) {
        X[(m0 + r + chi) * LDK + nt * 16 + am] = fmaxf(acc[nt][r] + bb, 0.0f);
      }
    }
  }

  // ---- layer 3: (128x64) x (64x128), fused ReLU + row-max ------------
  {
    v8f acc[8] = {};
    for (int k0 = 0; k0 < 64; k0 += 4) {
      const v2f a = Afrag(k0);
#pragma unroll
      for (int nt = 0; nt < 8; ++nt) {
        const v2f bf = Bfrag(W3, 128, k0, nt * 16, 63);
        acc[nt] = __builtin_amdgcn_wmma_f32_16x16x4_f32(false, a, false, bf,
                                                        (short)0, acc[nt], false, false);
      }
    }
#pragma unroll
    for (int nt = 0; nt < 8; ++nt) {
      const float bb = b3[nt * 16 + am];
      float mx = 0.0f;                       // ReLU output lower bound
#pragma unroll
      for (int r = 0; r < 8; ++r) mx = fmaxf(mx, fmaxf(acc[nt][r] + bb, 0.0f));
      mx = fmaxf(mx, __shfl_xor(mx, 16, 32));   // combine the two 8-row halves
      if (lane < 16) pmax[wave * 128 + nt * 16 + lane] = mx;
    }
  }
  __syncthreads();

  // combine the two waves covering each centroid's 32 samples
  for (int e = tid; e < 4 * 128; e += 256) {
    const int c = e >> 7, ch = e & 127;
    const float v = fmaxf(pmax[(2 * c) * 128 + ch], pmax[(2 * c + 1) * 128 + ch]);
    out_pts[(size_t)(s0 + c) * 128 + ch] = v;
  }
}

// =====================================================================
extern "C" void kernel_launch(void* const* d_in, const int* in_sizes, int n_in,
                              void* d_out, int out_size, void* d_ws, size_t ws_size,
                              hipStream_t stream) {
  const float* xyz = (const float*)d_in[0];
  const float* pts = (const float*)d_in[1];
  const float* W1  = (const float*)d_in[2];
  const float* b1  = (const float*)d_in[3];
  const float* W2  = (const float*)d_in[4];
  const float* b2  = (const float*)d_in[5];
  const float* W3  = (const float*)d_in[6];
  const float* b3  = (const float*)d_in[7];

  float* out        = (float*)d_out;
  float* new_xyz    = out;                              // B*NPOINT*3
  float* new_points = out + (size_t)BB * NPOINT * 3;    // B*NPOINT*128
  float* idx_out    = new_points + (size_t)BB * NPOINT * 128;  // B*NPOINT*32

  int* fps_idx = (int*)d_ws;                 // B*NPOINT ints
  int* idx_ws  = fps_idx + BB * NPOINT;      // B*NPOINT*NSAMPLE ints

  fps_kernel<<<BB, 256, 0, stream>>>(xyz, fps_idx, new_xyz);
  ball_kernel<<<(BB * NPOINT) / 8, 256, 0, stream>>>(xyz, new_xyz, fps_idx,
                                                     idx_ws, idx_out);
  mlp_kernel<<<(BB * NPOINT) / 4, 256, 0, stream>>>(xyz, pts, idx_ws, new_xyz,
                                                    W1, b1, W2, b2, W3, b3,
                                                    new_points);
}